// MemoryEfficientDiceLoss_1778116461275
// MI455X (gfx1250) — compile-verified
//
#include <hip/hip_runtime.h>

#define NCLS       62
#define TILE       128          // pixels per tile == threads per block (4 wave32)
#define HW         (512 * 512)  // 2^18
#define NPIX       (8 * HW)
#define NTILES     (NPIX / TILE)
#define GRID       2048
#define IGNORE_IDX 255
#define SMOOTH     1e-6f

// ---- CDNA5 async global->LDS copy path (gfx1250), with portable fallback ----
#if defined(__gfx1250__) && __has_builtin(__builtin_amdgcn_global_load_async_to_lds_b32) && \
    __has_builtin(__builtin_amdgcn_s_wait_asynccnt)
#define USE_ASYNC 1
#define WAIT_ASYNC(n) __builtin_amdgcn_s_wait_asynccnt(n)
typedef __attribute__((address_space(1))) int glob_i32_t;  // global (device) int
typedef __attribute__((address_space(3))) int lds_i32_t;   // LDS int
#else
#define USE_ASYNC 0
#define WAIT_ASYNC(n)
#endif

// Copy one pixel-column (62 channels, stride HW in global / stride TILE in LDS).
__device__ __forceinline__ void issue_tile_copy(const float* __restrict__ gcol,
                                                float* __restrict__ lcol) {
#if USE_ASYNC
#pragma unroll
  for (int c = 0; c < NCLS; ++c) {
    __builtin_amdgcn_global_load_async_to_lds_b32(
        (glob_i32_t*)(gcol + (size_t)c * HW),
        (lds_i32_t*)(lcol + c * TILE),
        /*offset=*/0, /*cpol=*/0);
  }
#else
#pragma unroll
  for (int c = 0; c < NCLS; ++c) lcol[c * TILE] = gcol[(size_t)c * HW];
#endif
}

extern "C" __global__ void __launch_bounds__(TILE) dice_partial(
    const float* __restrict__ pred, const int* __restrict__ target,
    float* __restrict__ ws) {
  // double-buffered staged tile: [buf][channel][pixel-in-tile]
  __shared__ float stage[2][NCLS * TILE];   // 63488 B
  __shared__ float s_pred[NCLS];
  __shared__ float s_inter[NCLS];
  __shared__ float s_tgt[NCLS];

  const int tid = threadIdx.x;
  for (int i = tid; i < NCLS; i += TILE) {
    s_pred[i] = 0.f; s_inter[i] = 0.f; s_tgt[i] = 0.f;
  }
  __syncthreads();

  float acc[NCLS];                // per-thread per-class softmax-prob sums
#pragma unroll
  for (int c = 0; c < NCLS; ++c) acc[c] = 0.f;

  const int tile0 = blockIdx.x;
  const int tstep = gridDim.x;

  // global pointer to channel-0 of this thread's pixel for tile t
  // pixel p = t*TILE + tid ; n = p / HW ; addr = p + n*(C-1)*HW
  auto gcol_for_tile = [&](int t) -> const float* {
    int p = t * TILE + tid;
    int n = p >> 18;  // HW == 2^18 ; tiles never straddle n (HW % TILE == 0)
    return pred + (size_t)p + (size_t)n * (size_t)(NCLS - 1) * (size_t)HW;
  };

  int buf = 0;
  if (tile0 < NTILES) issue_tile_copy(gcol_for_tile(tile0), &stage[0][tid]);

  for (int t = tile0; t < NTILES; t += tstep) {
    const int tn = t + tstep;
    if (tn < NTILES) {
      issue_tile_copy(gcol_for_tile(tn), &stage[buf ^ 1][tid]);
      WAIT_ASYNC(NCLS);  // in-order completion: <=62 outstanding => tile t resident
    } else {
      WAIT_ASYNC(0);
    }

    const float* col = &stage[buf][tid];
    const int p = t * TILE + tid;
    const int tg = target[p];
    const bool valid = (tg != IGNORE_IDX);
    const int ts = valid ? tg : 0;

    // pass 1: channel max
    float m = -3.402823466e+38f;
#pragma unroll
    for (int c = 0; c < NCLS; ++c) m = fmaxf(m, col[c * TILE]);

    // pass 2: exp-sum
    float s = 0.f;
#pragma unroll
    for (int c = 0; c < NCLS; ++c) s += __expf(col[c * TILE] - m);
    const float w = valid ? (1.0f / s) : 0.0f;  // folds the valid mask into the scale

    // pass 3: accumulate per-class prob sums, capture target-class numerator
    float et = 0.f;
#pragma unroll
    for (int c = 0; c < NCLS; ++c) {
      const float e = __expf(col[c * TILE] - m);
      acc[c] = fmaf(e, w, acc[c]);
      if (c == ts) et = e;  // v_cndmask, no memory gather
    }
    atomicAdd(&s_inter[ts], et * w);
    atomicAdd(&s_tgt[ts], valid ? 1.0f : 0.0f);

    buf ^= 1;
  }

  // wave32 reduce each class accumulator, one LDS atomic per wave per class
#pragma unroll
  for (int c = 0; c < NCLS; ++c) {
    float v = acc[c];
    v += __shfl_xor(v, 16, 32);
    v += __shfl_xor(v, 8, 32);
    v += __shfl_xor(v, 4, 32);
    v += __shfl_xor(v, 2, 32);
    v += __shfl_xor(v, 1, 32);
    if ((tid & 31) == 0) atomicAdd(&s_pred[c], v);
  }
  __syncthreads();

  for (int i = tid; i < NCLS; i += TILE) {
    atomicAdd(&ws[i], s_pred[i]);
    atomicAdd(&ws[NCLS + i], s_inter[i]);
    atomicAdd(&ws[2 * NCLS + i], s_tgt[i]);
  }
}

extern "C" __global__ void dice_init(float* __restrict__ ws) {
  const int i = threadIdx.x;
  if (i < 3 * NCLS) ws[i] = 0.f;
}

extern "C" __global__ void dice_final(const float* __restrict__ ws,
                                      float* __restrict__ out) {
  const int tid = threadIdx.x;  // 32 threads, 1 wave
  float dsum = 0.f, cnt = 0.f;
  for (int c = tid; c < NCLS; c += 32) {
    const float ps = ws[c];
    const float it = ws[NCLS + c];
    const float tg = ws[2 * NCLS + c];
    const float un = ps + tg;
    const float dice = (2.f * it + SMOOTH) / (un + SMOOTH);
    if (un > 0.f) { dsum += dice; cnt += 1.f; }
  }
  dsum += __shfl_xor(dsum, 16, 32);
  dsum += __shfl_xor(dsum, 8, 32);
  dsum += __shfl_xor(dsum, 4, 32);
  dsum += __shfl_xor(dsum, 2, 32);
  dsum += __shfl_xor(dsum, 1, 32);
  cnt += __shfl_xor(cnt, 16, 32);
  cnt += __shfl_xor(cnt, 8, 32);
  cnt += __shfl_xor(cnt, 4, 32);
  cnt += __shfl_xor(cnt, 2, 32);
  cnt += __shfl_xor(cnt, 1, 32);
  if (tid == 0) {
    const float mean = (cnt > 0.f) ? (dsum / cnt) : 1.0f;
    out[0] = 1.0f - mean;
  }
}

extern "C" void kernel_launch(void* const* d_in, const int* in_sizes, int n_in,
                              void* d_out, int out_size, void* d_ws, size_t ws_size,
                              hipStream_t stream) {
  const float* pred = (const float*)d_in[0];   // (8, 62, 512, 512) fp32
  const int* target = (const int*)d_in[1];     // (8, 512, 512) int32
  float* ws = (float*)d_ws;                    // 3*62 float accumulators
  float* out = (float*)d_out;                  // scalar fp32 loss

  dice_init<<<1, 192, 0, stream>>>(ws);
  dice_partial<<<GRID, TILE, 0, stream>>>(pred, target, ws);
  dice_final<<<1, 32, 0, stream>>>(ws, out);
}